// ComputeRealAngleInput_81827716923458
// MI455X (gfx1250) — compile-verified
//
#include <hip/hip_runtime.h>
#include <math.h>

// CDNA5 / gfx1250. wave32.
typedef __attribute__((ext_vector_type(2))) float v2f;
typedef __attribute__((ext_vector_type(8))) float v8f;

#define NNEIGH 16
#define NFEAT  64
#define NPAIR  (NNEIGH * (NNEIGH - 1))   // 240
#define ROWLEN (3 + 3 * NFEAT)           // 195
#define CHUNK  (NPAIR * ROWLEN)          // 46800 floats per center

__global__ __launch_bounds__(256) void angdesc_kernel(
    const int*   __restrict__ atom_i_idx,   // [C]
    const int*   __restrict__ atom_j_idx,   // [C,16]
    const float* __restrict__ dist_ij,      // [C,16]
    const float* __restrict__ atoms_xyz,    // [A,3]
    const int*   __restrict__ atoms_long,   // [A,2]
    const float* __restrict__ embed_table,  // [S,64]
    float*       __restrict__ out,          // [C ints][C*240*195 f32]
    int C)
{
    const int c   = blockIdx.x;
    const int tid = threadIdx.x;

    __shared__ float s_vec[NNEIGH][4];     // vec_ij (xyz, pad 0)
    __shared__ float s_rnorm[NNEIGH];      // 1/max(|vec|, eps)
    __shared__ float s_dist[NNEIGH];
    __shared__ float s_rdist[NNEIGH];
    __shared__ int   s_specj[NNEIGH];
    __shared__ float s_embI[NFEAT];        // emb_i
    __shared__ float s_embS[NNEIGH * NFEAT]; // emb_j[n]/dist[n]
    __shared__ float s_ang[NNEIGH * NNEIGH]; // angles 16x16

    const int ii = atom_i_idx[c];          // uniform -> scalar load
    const float xi = atoms_xyz[ii * 3 + 0];
    const float yi = atoms_xyz[ii * 3 + 1];
    const float zi = atoms_xyz[ii * 3 + 2];

    // ---- Phase 1: per-neighbor gathers + emb_i ----
    if (tid < NNEIGH) {
        const int jj = atom_j_idx[c * NNEIGH + tid];
        const float dx = atoms_xyz[jj * 3 + 0] - xi;
        const float dy = atoms_xyz[jj * 3 + 1] - yi;
        const float dz = atoms_xyz[jj * 3 + 2] - zi;
        s_vec[tid][0] = dx; s_vec[tid][1] = dy; s_vec[tid][2] = dz; s_vec[tid][3] = 0.0f;
        const float nrm = sqrtf(dx * dx + dy * dy + dz * dz);
        s_rnorm[tid] = 1.0f / fmaxf(nrm, 1e-8f);
        const float d = dist_ij[c * NNEIGH + tid];
        s_dist[tid]  = d;
        s_rdist[tid] = 1.0f / d;
        s_specj[tid] = atoms_long[jj * 2 + 1];
    } else if (tid >= 32 && tid < 32 + NFEAT) {
        const int si = atoms_long[ii * 2 + 1];
        s_embI[tid - 32] = embed_table[si * NFEAT + (tid - 32)];
    }
    __syncthreads();

    // ---- Phase 2a (wave 0 only; tid<32 is wave-uniform -> EXEC all ones):
    // dot[j,k] = vec[j].vec[k] via V_WMMA_F32_16X16X4_F32 with B == A^T.
    // A 16x4 layout: VGPR0 = {K0 | K2}, VGPR1 = {K1 | K3} across lane halves;
    // for B = A^T (4x16) the lane layout is identical, so feed `a` twice.
    if (tid < 32) {
        const int  m  = tid & 15;
        const bool hi = tid >= 16;
        v2f a;
        a.x = hi ? s_vec[m][2] : s_vec[m][0];   // K2=z | K0=x
        a.y = hi ? 0.0f        : s_vec[m][1];   // K3=0 | K1=y
        v8f acc = {0.f, 0.f, 0.f, 0.f, 0.f, 0.f, 0.f, 0.f};
        v8f dmat = __builtin_amdgcn_wmma_f32_16x16x4_f32(
            /*neg_a=*/false, a, /*neg_b=*/false, a,
            /*c_mod=*/(short)0, acc, /*reuse_a=*/false, /*reuse_b=*/false);
        // D layout: VGPR r -> (M=r, N=lane) lanes 0-15; (M=r+8, N=lane-16) lanes 16-31
        const float rn_col = s_rnorm[m];
        const int mofs = hi ? 8 : 0;
#pragma unroll
        for (int r = 0; r < 8; ++r) {
            const int M = r + mofs;
            const float cosv = dmat[r] * s_rnorm[M] * rn_col;
            s_ang[M * NNEIGH + m] = acosf(cosv * 0.9999f);
        }
    }

    // ---- Phase 2b (all waves): scaled neighbor embeddings, emb_j[n]/d[n] ----
    for (int idx = tid; idx < NNEIGH * NFEAT; idx += 256) {
        const int n = idx >> 6, f = idx & 63;
        s_embS[idx] = embed_table[s_specj[n] * NFEAT + f] * s_rdist[n];
    }
    __syncthreads();

    // ---- Phase 3: stream output (bandwidth-bound part) ----
    if (tid == 0) ((int*)out)[c] = ii;   // int32 header, bit-exact
    float* chunk = out + C + (size_t)c * CHUNK;

    // headers: 240 pairs x 3 (d_j, d_k, ang)
    for (int idx = tid; idx < NPAIR * 3; idx += 256) {
        const int p = idx / 3, f = idx - p * 3;
        const int j = p / 15, r = p - j * 15;
        const int k = r + (r >= j ? 1 : 0);
        const float v = (f == 0) ? s_dist[j]
                      : (f == 1) ? s_dist[k]
                                 : s_ang[j * NNEIGH + k];
        chunk[p * ROWLEN + f] = v;
    }
    // emb_i block (same 64 values every row)
    for (int idx = tid; idx < NPAIR * NFEAT; idx += 256) {
        const int p = idx >> 6, f = idx & 63;
        chunk[p * ROWLEN + 3 + f] = s_embI[f];
    }
    // emb_j / d_j block
    for (int idx = tid; idx < NPAIR * NFEAT; idx += 256) {
        const int p = idx >> 6, f = idx & 63;
        const int j = p / 15;
        chunk[p * ROWLEN + 3 + NFEAT + f] = s_embS[(j << 6) + f];
    }
    // emb_k / d_k block
    for (int idx = tid; idx < NPAIR * NFEAT; idx += 256) {
        const int p = idx >> 6, f = idx & 63;
        const int j = p / 15, r = p - j * 15;
        const int k = r + (r >= j ? 1 : 0);
        chunk[p * ROWLEN + 3 + 2 * NFEAT + f] = s_embS[(k << 6) + f];
    }
}

extern "C" void kernel_launch(void* const* d_in, const int* in_sizes, int n_in,
                              void* d_out, int out_size, void* d_ws, size_t ws_size,
                              hipStream_t stream) {
    // setup_inputs order: 0=nNeigh(scalar), 1=atom_i_idx, 2=atom_j_idx,
    // 3=dist_ij, 4=atoms_xyz, 5=atoms_long, 6=embed_table
    const int*   atom_i = (const int*)  d_in[1];
    const int*   atom_j = (const int*)  d_in[2];
    const float* dist   = (const float*)d_in[3];
    const float* xyz    = (const float*)d_in[4];
    const int*   along  = (const int*)  d_in[5];
    const float* emb    = (const float*)d_in[6];
    float* out = (float*)d_out;
    const int C = in_sizes[1];

    hipLaunchKernelGGL(angdesc_kernel, dim3(C), dim3(256), 0, stream,
                       atom_i, atom_j, dist, xyz, along, emb, out, C);
}